// SO3FlowSampler_438086664547
// MI455X (gfx1250) — compile-verified
//
#include <hip/hip_runtime.h>
#include <stdint.h>

// ---------------------------------------------------------------------------
// SO(3) flow sampling, MI455X (gfx1250).
// HBM-bound (≈2.7 FLOP/byte): optimize the data path, not the math.
//  - TDM (tensor_load_to_lds) bulk-DMAs each block's contiguous input tiles
//    into LDS (TENSORcnt + s_wait_tensorcnt + workgroup barrier).
//  - wave32-sized blocks (256 thr = 8 waves), conflict-free LDS reads.
//  - Direct coalesced stores; RT hints keep the 149MB set L2-resident (192MB).
// ---------------------------------------------------------------------------

#define BLK 256
// LDS layout (dynamic shared memory starts at LDS byte offset 0):
//   [0,      9216)  x0 tile : BLK*9 floats
//   [9216,  18432)  x1 tile : BLK*9 floats
//   [18432, 19456)  t  tile : BLK floats
//   [19456, 19712)  mask    : BLK bytes
#define OFF_X0   0u
#define OFF_X1   9216u
#define OFF_T    18432u
#define OFF_MASK 19456u
#define SMEM_BYTES 19712

typedef unsigned int u32x4 __attribute__((ext_vector_type(4)));
typedef unsigned int u32x8 __attribute__((ext_vector_type(8)));

// Issue one TDM descriptor copying `nelem` contiguous elements (element size
// 1<<dsz bytes) from global `gaddr` to LDS byte offset `lds_off`.
// D# group layouts per CDNA5 ISA §8.3 (group0) / §8.4 (group1); groups 2/3
// zeroed (tile_dim2..4 = 0 => unused).
__device__ __forceinline__ void tdm_load_1d(uint64_t gaddr, uint32_t lds_off,
                                            uint32_t nelem, uint32_t dsz) {
  u32x4 g0;
  g0.x = 1u;                                   // count=1 (valid descriptor)
  g0.y = lds_off;                              // lds_addr
  g0.z = (uint32_t)gaddr;                      // global_addr[31:0]
  g0.w = (uint32_t)(gaddr >> 32) | (2u << 30); // global_addr[56:32] | type=2

  u32x8 g1;
  g1[0] = dsz << 16;                           // workgroup_mask=0, data_size
  g1[1] = (nelem & 0xFFFFu) << 16;             // tensor_dim0[15:0]
  g1[2] = ((nelem >> 16) & 0xFFFFu)            // tensor_dim0[31:16]
        | (1u << 16);                          // tensor_dim1 = 1
  g1[3] = (nelem & 0xFFFFu) << 16;             // tile_dim0 = nelem (<=65535)
  g1[4] = 1u;                                  // tile_dim1 = 1, tile_dim2 = 0
  g1[5] = nelem;                               // tensor_dim0_stride[31:0]
  g1[6] = 0u;                                  // stride hi, tensor_dim1_stride lo
  g1[7] = 0u;

  u32x4 g2 = {0u, 0u, 0u, 0u};
  u32x4 g3 = {0u, 0u, 0u, 0u};

  asm volatile("tensor_load_to_lds %0, %1, %2, %3"
               :
               : "s"(g0), "s"(g1), "s"(g2), "s"(g3)
               : "memory");
}

// log_so3: rotation matrix C (row-major 9) -> axis*angle w[3]
__device__ __forceinline__ void log_so3_dev(const float* C, float* w) {
  float tr = C[0] + C[4] + C[8];
  float c  = 0.5f * (tr - 1.0f);
  c = fminf(1.0f, fmaxf(-1.0f, c));
  float wx = 0.5f * (C[7] - C[5]);
  float wy = 0.5f * (C[2] - C[6]);
  float wz = 0.5f * (C[3] - C[1]);
  float sn2 = wx * wx + wy * wy + wz * wz;
  float sn  = sqrtf(fmaxf(sn2, 1e-12f));
  float th  = atan2f(sn, c);
  float s   = (sn2 < 1e-8f) ? 1.0f : th / sn;
  w[0] = s * wx; w[1] = s * wy; w[2] = s * wz;
}

// exp_so3: v[3] -> rotation E (row-major 9), matching reference branches
__device__ __forceinline__ void exp_so3_dev(const float* v, float* E) {
  float x = v[0], y = v[1], z = v[2];
  float t2  = x * x + y * y + z * z;
  float t2s = fmaxf(t2, 1e-12f);
  float th  = sqrtf(t2s);
  float a, b;
  if (t2 < 1e-8f) {
    a = 1.0f - t2 * (1.0f / 6.0f);
    b = 0.5f - t2 * (1.0f / 24.0f);
  } else {
    a = sinf(th) / th;
    b = (1.0f - cosf(th)) / t2s;
  }
  float xy = x * y, xz = x * z, yz = y * z;
  float KK00 = -(z * z + y * y);
  float KK11 = -(z * z + x * x);
  float KK22 = -(y * y + x * x);
  E[0] = 1.0f + b * KK00; E[1] = -a * z + b * xy; E[2] =  a * y + b * xz;
  E[3] =  a * z + b * xy; E[4] = 1.0f + b * KK11; E[5] = -a * x + b * yz;
  E[6] = -a * y + b * xz; E[7] =  a * x + b * yz; E[8] = 1.0f + b * KK22;
}

__global__ __launch_bounds__(BLK) void so3_flow_sampler_kernel(
    const float* __restrict__ x1g, const float* __restrict__ x0g,
    const float* __restrict__ tg, const unsigned char* __restrict__ mg,
    float* __restrict__ out_ut, float* __restrict__ out_xt, int n) {
  extern __shared__ unsigned char smem[];
  float* sx0        = (float*)(smem + OFF_X0);
  float* sx1        = (float*)(smem + OFF_X1);
  float* stt        = (float*)(smem + OFF_T);
  unsigned char* sm = (unsigned char*)(smem + OFF_MASK);

  const int base = blockIdx.x * BLK;
  int rem = n - base;
  if (rem > BLK) rem = BLK;

  // Wave 0 issues the four TDM descriptors for this block's tiles, then
  // drains TENSORcnt. All waves then sync so LDS contents are visible.
  if (threadIdx.x < 32) {
    tdm_load_1d((uint64_t)(uintptr_t)(x0g + (size_t)base * 9), OFF_X0,
                (uint32_t)rem * 9u, 2u);
    tdm_load_1d((uint64_t)(uintptr_t)(x1g + (size_t)base * 9), OFF_X1,
                (uint32_t)rem * 9u, 2u);
    tdm_load_1d((uint64_t)(uintptr_t)(tg + base), OFF_T, (uint32_t)rem, 2u);
    tdm_load_1d((uint64_t)(uintptr_t)(mg + base), OFF_MASK, (uint32_t)rem, 0u);
    __builtin_amdgcn_s_wait_tensorcnt(0);
  }
  __syncthreads();

  const int lt  = threadIdx.x;
  const int idx = base + lt;
  if (idx >= n) return;

  float R0[9], R1[9];
#pragma unroll
  for (int k = 0; k < 9; ++k) {
    R0[k] = sx0[lt * 9 + k];
    R1[k] = sx1[lt * 9 + k];
  }
  const float tt  = stt[lt];
  const bool  msk = sm[lt] != 0;
  if (!msk) {
#pragma unroll
    for (int k = 0; k < 9; ++k) R0[k] = R1[k];
  }

  // C = R0^T @ R1
  float C[9];
#pragma unroll
  for (int i = 0; i < 3; ++i)
#pragma unroll
    for (int j = 0; j < 3; ++j)
      C[i * 3 + j] = R0[0 * 3 + i] * R1[0 * 3 + j] +
                     R0[1 * 3 + i] * R1[1 * 3 + j] +
                     R0[2 * 3 + i] * R1[2 * 3 + j];

  float logv[3];
  log_so3_dev(C, logv);

  // E = exp_so3(t * logv);  xt = R0 @ E
  float v[3] = {tt * logv[0], tt * logv[1], tt * logv[2]};
  float E[9];
  exp_so3_dev(v, E);

  float XT[9];
#pragma unroll
  for (int i = 0; i < 3; ++i)
#pragma unroll
    for (int j = 0; j < 3; ++j)
      XT[i * 3 + j] = R0[i * 3 + 0] * E[0 * 3 + j] +
                      R0[i * 3 + 1] * E[1 * 3 + j] +
                      R0[i * 3 + 2] * E[2 * 3 + j];

  // delta = R0^T @ xt ;  w2 = log_so3(delta)
  float D[9];
#pragma unroll
  for (int i = 0; i < 3; ++i)
#pragma unroll
    for (int j = 0; j < 3; ++j)
      D[i * 3 + j] = R0[0 * 3 + i] * XT[0 * 3 + j] +
                     R0[1 * 3 + i] * XT[1 * 3 + j] +
                     R0[2 * 3 + i] * XT[2 * 3 + j];

  float w2[3];
  log_so3_dev(D, w2);
  const float x2 = w2[0], y2 = w2[1], z2 = w2[2];
  const float inv_t = 1.0f / tt;

  // ut = (xt @ hat(w2)) / t
  float UT[9];
#pragma unroll
  for (int i = 0; i < 3; ++i) {
    const float m0 = XT[i * 3 + 0], m1 = XT[i * 3 + 1], m2 = XT[i * 3 + 2];
    UT[i * 3 + 0] = (m1 * z2 - m2 * y2) * inv_t;
    UT[i * 3 + 1] = (m2 * x2 - m0 * z2) * inv_t;
    UT[i * 3 + 2] = (m0 * y2 - m1 * x2) * inv_t;
  }

  float* pu = out_ut + (size_t)idx * 9;
  float* px = out_xt + (size_t)idx * 9;
  if (msk) {
#pragma unroll
    for (int k = 0; k < 9; ++k) { pu[k] = UT[k]; px[k] = XT[k]; }
  } else {
#pragma unroll
    for (int k = 0; k < 9; ++k) { pu[k] = 0.0f; px[k] = R1[k]; }
  }
}

extern "C" void kernel_launch(void* const* d_in, const int* in_sizes, int n_in,
                              void* d_out, int out_size, void* d_ws, size_t ws_size,
                              hipStream_t stream) {
  (void)n_in; (void)out_size; (void)d_ws; (void)ws_size;
  // setup_inputs order: x1 [B*9] f32, x0 [B*9] f32, t [B] f32, mask [B] bool(u8)
  const float*         x1   = (const float*)d_in[0];
  const float*         x0   = (const float*)d_in[1];
  const float*         t    = (const float*)d_in[2];
  const unsigned char* mask = (const unsigned char*)d_in[3];
  const int n = in_sizes[2];

  float* out    = (float*)d_out;
  float* out_ut = out;                    // first output: ut [B*9]
  float* out_xt = out + (size_t)n * 9;    // second output: xt [B*9]

  const int grid = (n + BLK - 1) / BLK;
  so3_flow_sampler_kernel<<<grid, BLK, SMEM_BYTES, stream>>>(
      x1, x0, t, mask, out_ut, out_xt, n);
}